// OCRTrainNet_10247791969020
// MI455X (gfx1250) — compile-verified
//
#include <hip/hip_runtime.h>
#include <math.h>

typedef float v2f __attribute__((ext_vector_type(2)));
typedef float v8f __attribute__((ext_vector_type(8)));

#define EPSF 1e-7f

// ---------------------------------------------------------------------------
// Kernel 1: fused 3x3 conv (implicit GEMM via V_WMMA_F32_16X16X4_F32) + losses
// Grid: 32*128 blocks (one image-row each), 256 threads = 8 waves.
// Each wave computes a 16(pixel) x 16(outch, 5 real) tile, K = 9*128.
// All inner-loop loads are unconditional (no exec diamonds): OOB pixels use a
// clamped address and are zeroed by multiplying A by okf; out-channel padding
// uses a zero slot (n==5) in the LDS weight-fragment table.
// ---------------------------------------------------------------------------
__global__ __launch_bounds__(256)
void conv_loss_kernel(const float* __restrict__ fmap,
                      const float* __restrict__ bbox_true,
                      const float* __restrict__ Wc,
                      const float* __restrict__ bc,
                      float* __restrict__ ws)
{
    // B-fragment table: sBf[(khw*64 + ci2)*6 + n] = (W[khw][2*ci2][n], W[khw][2*ci2+1][n])
    // n==5 slot is zero (pads out-channels 5..15).
    __shared__ v2f   sBf[9 * 64 * 6];        // 27648 B
    __shared__ float sBias[5];
    __shared__ float sTile[8][16][17];       // per-wave D tile, padded stride

    const int tid = threadIdx.x;
    for (int i = tid; i < 9 * 64 * 6; i += 256) {
        const int khw = i / 384;
        const int rem = i % 384;
        const int ci2 = rem / 6;
        const int n   = rem % 6;
        v2f val;
        if (n < 5) {
            val.x = Wc[((khw * 128) + 2 * ci2)     * 5 + n];
            val.y = Wc[((khw * 128) + 2 * ci2 + 1) * 5 + n];
        } else {
            val.x = 0.f; val.y = 0.f;
        }
        sBf[i] = val;
    }
    if (tid < 5) sBias[tid] = bc[tid];
    __syncthreads();

    const int n    = blockIdx.x >> 7;        // image
    const int h    = blockIdx.x & 127;       // output row
    const int wave = tid >> 5;
    const int lane = tid & 31;
    const int w0   = wave * 16;              // first pixel of this wave's tile
    const int m    = lane & 15;              // M (A) / N (B,C,D) index
    const int half = lane >> 4;              // 0 or 1
    const int nclamp = (m < 5) ? m : 5;      // zero slot for padded out-channels

    v8f acc = {0.f, 0.f, 0.f, 0.f, 0.f, 0.f, 0.f, 0.f};

    for (int kh = 0; kh < 3; ++kh) {
        const int y  = h + kh - 1;
        const int yc = (y < 0) ? 0 : ((y > 127) ? 127 : y);
        for (int kw = 0; kw < 3; ++kw) {
            const int x  = w0 + m + kw - 1;
            const int xc = (x < 0) ? 0 : ((x > 127) ? 127 : x);
            const float okf = ((y >= 0) && (y < 128) && (x >= 0) && (x < 128)) ? 1.f : 0.f;
            // A source: fmap[n][yc][xc][4t + 2*half .. +1] as v2f, always valid
            const v2f* src = (const v2f*)(fmap + (((size_t)n * 128 + yc) * 128 + xc) * 128) + half;
            // B source: fragment table, stride 12 v2f per K-chunk of 4
            const v2f* wsrc = &sBf[(((kh * 3 + kw) * 64) + half) * 6 + nclamp];
            #pragma unroll 8
            for (int t = 0; t < 32; ++t) {
                v2f a = src[2 * t];
                a.x *= okf; a.y *= okf;
                const v2f b = wsrc[t * 12];
                acc = __builtin_amdgcn_wmma_f32_16x16x4_f32(
                    false, a, false, b, (short)0, acc, false, false);
            }
        }
    }

    // D layout: VGPR v -> M = v + 8*half, N = m
    #pragma unroll
    for (int v = 0; v < 8; ++v)
        sTile[wave][v + 8 * half][m] = acc[v];
    __syncthreads();

    // Loss contributions: lanes 0..15 each own pixel (n, h, w0+lane)
    float c_conf = 0.f, c_m = 0.f, c_iou = 0.f, c_pos = 0.f;
    if (lane < 16) {
        const int w = w0 + lane;
        const size_t pix = ((size_t)n * 128 + h) * 128 + w;
        const float* yt = bbox_true + pix * 5;
        float yp[5];
        #pragma unroll
        for (int c = 0; c < 5; ++c) yp[c] = sTile[wave][lane][c] + sBias[c];

        const float conf_t = yt[0];
        const float p = 1.f / (1.f + expf(-yp[0]));
        if (conf_t == 1.0f) {
            const float om = 1.f - p;
            c_conf = -(0.25f * om * om * logf(p + EPSF));
        } else if (conf_t == 0.0f) {
            c_conf = -(0.75f * p * p * logf(1.f - p + EPSF));
        }
        c_m = (conf_t != -1.0f) ? 1.f : 0.f;
        if (conf_t == 1.0f) {
            const float area_t = (yt[3] + yt[1]) * (yt[4] + yt[2]);
            const float area_p = fmaxf((yp[3] + yp[1]) * (yp[4] + yp[2]), 0.f);
            const float xi = fmaxf(fminf(yt[3], yp[3]) + fminf(yt[1], yp[1]), 0.f);
            const float yi = fmaxf(fminf(yt[4], yp[4]) + fminf(yt[2], yp[2]), 0.f);
            const float inter = xi * yi;
            const float iou = inter / (area_t + area_p - inter + EPSF);
            c_iou = logf(iou + EPSF);
            c_pos = 1.f;
        }
    }
    // wave32 butterfly reduction
    #pragma unroll
    for (int off = 16; off >= 1; off >>= 1) {
        c_conf += __shfl_xor(c_conf, off, 32);
        c_m    += __shfl_xor(c_m,    off, 32);
        c_iou  += __shfl_xor(c_iou,  off, 32);
        c_pos  += __shfl_xor(c_pos,  off, 32);
    }
    if (lane == 0) {
        atomicAdd(&ws[0], c_conf);
        atomicAdd(&ws[1], c_m);
        atomicAdd(&ws[2], c_iou);
        atomicAdd(&ws[3], c_pos);
    }
}

__global__ void init_ws(float* ws)
{
    if (threadIdx.x < 4) ws[threadIdx.x] = 0.f;
}

__global__ void finalize_loss(const float* __restrict__ ws, float* __restrict__ out)
{
    if (threadIdx.x == 0) {
        out[0] = ws[0] / fmaxf(ws[1], 1.f);   // loss_conf
        out[1] = -ws[2] / fmaxf(ws[3], 1.f);  // loss_iou
    }
}

// ---------------------------------------------------------------------------
// ROI pooling (memory-bound gathers)
// ---------------------------------------------------------------------------
struct PoolParams {
    float bx0, by0, map_w, map_h;
    int   long_len;
    bool  nz;
};

__device__ __forceinline__ PoolParams pool_params(const float* box, bool horizontal)
{
    float b0 = box[0], b1 = box[1], b2 = box[2], b3 = box[3];
    const float bw0 = b2 - b0, bh0 = b3 - b1;
    const bool keep = horizontal ? (bw0 >= bh0) : (bw0 < bh0);
    if (!keep) { b0 = b1 = b2 = b3 = 0.f; }
    const float bw = b2 - b0, bh = b3 - b1;
    const bool nz = (bw >= 0.1f) || (bh >= 0.1f);
    const float bww = (bw <= 0.f) ? 1e-4f : bw;
    const float bhh = (bh <= 0.f) ? 1e-4f : bh;
    const float ratio = nz ? (horizontal ? bww / bhh : bhh / bww) : 0.f;
    int long_len = (int)ceilf(ratio * 8.0f);
    long_len = (long_len < 128) ? long_len : 128;
    const float lf = fmaxf((float)long_len, 2.0f);
    PoolParams p;
    p.nz = nz; p.long_len = long_len;
    p.bx0 = b0; p.by0 = b1;
    if (horizontal) { p.map_w = bw / (lf - 1.f); p.map_h = bh / 7.f; }
    else            { p.map_w = bw / 7.f;        p.map_h = bh / (lf - 1.f); }
    return p;
}

__device__ __forceinline__ void bil1d(float v, int& i0, int& i1, float& w0, float& w1)
{
    const int a = (int)floorf(v);
    int a1 = a + 1;
    a1 = (a1 < 0) ? 0 : ((a1 > 127) ? 127 : a1);
    int a0 = (a < 0) ? 0 : ((a > 127) ? 127 : a);
    w0 = (float)a1 - v;   // weight paired with i0 along this axis (ref semantics)
    w1 = v - (float)a0;   // weight paired with i1
    i0 = a0; i1 = a1;
}

// roi_h: out (32, 8, 128, 128); grid 32*8 blocks, 128 threads (channels)
__global__ __launch_bounds__(128)
void roi_h_kernel(const float* __restrict__ fmap, const float* __restrict__ boxes,
                  float* __restrict__ out_roi, float* __restrict__ out_w)
{
    const int b = blockIdx.x >> 3;
    const int r = blockIdx.x & 7;
    const int c = threadIdx.x;
    const PoolParams p = pool_params(boxes + b * 4, true);

    const float yv = r * p.map_h + p.by0;
    int iy0, iy1; float wy0, wy1;
    bil1d(yv, iy0, iy1, wy0, wy1);

    const float* img  = fmap + (size_t)b * 128 * 128 * 128;
    const float* row0 = img + (size_t)iy0 * 128 * 128;
    const float* row1 = img + (size_t)iy1 * 128 * 128;
    float* dst = out_roi + (((size_t)b * 8 + r) * 128) * 128;

    for (int x = 0; x < 128; ++x) {
        const float xv = x * p.map_w + p.bx0;
        int ix0, ix1; float wx0, wx1;
        bil1d(xv, ix0, ix1, wx0, wx1);
        const float Ia = row0[(size_t)ix0 * 128 + c];
        const float Ib = row0[(size_t)ix1 * 128 + c];
        const float Ic = row1[(size_t)ix0 * 128 + c];
        const float Id = row1[(size_t)ix1 * 128 + c];
        float val = Ia * (wx0 * wy0) + Ib * (wx0 * wy1)
                  + Ic * (wx1 * wy0) + Id * (wx1 * wy1);
        const float valid = ((x < p.long_len) && p.nz) ? 1.f : 0.f;
        dst[(size_t)x * 128 + c] = val * valid;
    }
    if (r == 0 && c == 0) out_w[b] = (float)p.long_len;
}

// roi_v: out (32, 128, 8, 128); grid 32*128 blocks, 128 threads (channels)
__global__ __launch_bounds__(128)
void roi_v_kernel(const float* __restrict__ fmap, const float* __restrict__ boxes,
                  float* __restrict__ out_roi, float* __restrict__ out_h)
{
    const int b    = blockIdx.x >> 7;
    const int yidx = blockIdx.x & 127;
    const int c    = threadIdx.x;
    const PoolParams p = pool_params(boxes + b * 4, false);

    const float yv = yidx * p.map_h + p.by0;
    int iy0, iy1; float wy0, wy1;
    bil1d(yv, iy0, iy1, wy0, wy1);

    const float* img  = fmap + (size_t)b * 128 * 128 * 128;
    const float* row0 = img + (size_t)iy0 * 128 * 128;
    const float* row1 = img + (size_t)iy1 * 128 * 128;
    float* dst = out_roi + (((size_t)b * 128 + yidx) * 8) * 128;
    const float validrow = ((yidx < p.long_len) && p.nz) ? 1.f : 0.f;

    for (int r = 0; r < 8; ++r) {
        const float xv = r * p.map_w + p.bx0;
        int ix0, ix1; float wx0, wx1;
        bil1d(xv, ix0, ix1, wx0, wx1);
        const float Ia = row0[(size_t)ix0 * 128 + c];
        const float Ib = row0[(size_t)ix1 * 128 + c];
        const float Ic = row1[(size_t)ix0 * 128 + c];
        const float Id = row1[(size_t)ix1 * 128 + c];
        float val = Ia * (wx0 * wy0) + Ib * (wx0 * wy1)
                  + Ic * (wx1 * wy0) + Id * (wx1 * wy1);
        dst[(size_t)r * 128 + c] = val * validrow;
    }
    if (yidx == 0 && c == 0) out_h[b] = (float)p.long_len;
}

// ---------------------------------------------------------------------------
extern "C" void kernel_launch(void* const* d_in, const int* in_sizes, int n_in,
                              void* d_out, int out_size, void* d_ws, size_t ws_size,
                              hipStream_t stream)
{
    const float* fmap      = (const float*)d_in[0];
    const float* boxes     = (const float*)d_in[1];
    const float* bbox_true = (const float*)d_in[2];
    const float* Wc        = (const float*)d_in[3];
    const float* bc        = (const float*)d_in[4];
    float* out = (float*)d_out;
    float* ws  = (float*)d_ws;

    const size_t ROIH_OFF = 2;
    const size_t W_OFF    = ROIH_OFF + (size_t)32 * 8 * 128 * 128;
    const size_t ROIV_OFF = W_OFF + 32;
    const size_t H_OFF    = ROIV_OFF + (size_t)32 * 128 * 8 * 128;

    init_ws<<<1, 32, 0, stream>>>(ws);
    conv_loss_kernel<<<32 * 128, 256, 0, stream>>>(fmap, bbox_true, Wc, bc, ws);
    finalize_loss<<<1, 32, 0, stream>>>(ws, out);
    roi_h_kernel<<<32 * 8,   128, 0, stream>>>(fmap, boxes, out + ROIH_OFF, out + W_OFF);
    roi_v_kernel<<<32 * 128, 128, 0, stream>>>(fmap, boxes, out + ROIV_OFF, out + H_OFF);
}